// CRFLayer_21500606284308
// MI455X (gfx1250) — compile-verified
//
#include <hip/hip_runtime.h>

typedef __attribute__((ext_vector_type(2))) float v2f;
typedef __attribute__((ext_vector_type(4))) float v4f;
typedef __attribute__((ext_vector_type(8))) float v8f;

#define S_LEN   128
#define BATCH   64
#define TAGS    256
#define NEGV    (-10000.0f)
#define LDS_P   17          // stride padding: 17 coprime with 64 banks
#define COLS    16          // batch columns per workgroup
#define NTHREAD 512         // 16 waves of 32

__global__ __launch_bounds__(NTHREAD)
void crf_fwd_wmma_kernel(const float* __restrict__ feats,
                         const float* __restrict__ mask,
                         const float* __restrict__ trans,
                         float* __restrict__ out)
{
    __shared__ float alphaS[TAGS * LDS_P];
    __shared__ float qS[TAGS * LDS_P];
    __shared__ float colMax[COLS];

    const int tid  = threadIdx.x;
    const int lane = tid & 31;
    const int w    = tid >> 5;            // wave id 0..15 -> owns tag rows [16w,16w+16)
    const int half = lane >> 4;           // 0: lanes 0-15, 1: lanes 16-31
    const int mrow = lane & 15;           // M index for A / N index for B/C
    const int c0   = blockIdx.x * COLS;   // batch column base for this workgroup

    // ---- A fragments: E = exp(transition), rows 16w..16w+15, all 256 K values.
    // 16x4 f32 A layout: lane = M (both halves); VGPR0: K=0 (lanes0-15)/K=2 (lanes16-31);
    // VGPR1: K=1 / K=3.  Held in registers for the whole kernel (E is step-invariant).
    v2f a[64];
    {
        const float* rowp = trans + (size_t)(16 * w + mrow) * TAGS + 2 * half;
#pragma unroll
        for (int k = 0; k < 64; ++k) {
            float2 t = *(const float2*)(rowp + 4 * k);
            v2f e; e.x = expf(t.x); e.y = expf(t.y);
            a[k] = e;
        }
    }

    // ---- init alpha: 0 at START_TAG(=0), NEG elsewhere
    for (int i = tid; i < TAGS * COLS; i += NTHREAD) {
        int tag = i >> 4, col = i & 15;
        alphaS[tag * LDS_P + col] = (tag == 0) ? 0.0f : NEGV;
    }
    __syncthreads();

    for (int s = 0; s < S_LEN; ++s) {
        // (ab) fused: wave w owns column w. Cache the column in registers,
        // butterfly max (all lanes get the result), write q = exp(alpha - max).
        // One barrier instead of two per step.
        {
            float av[8];
#pragma unroll
            for (int r = 0; r < 8; ++r)
                av[r] = alphaS[(lane + 32 * r) * LDS_P + w];
            float m = av[0];
#pragma unroll
            for (int r = 1; r < 8; ++r) m = fmaxf(m, av[r]);
#pragma unroll
            for (int off = 16; off > 0; off >>= 1)
                m = fmaxf(m, __shfl_xor(m, off, 32));
            if (lane == 0) colMax[w] = m;           // for phase (d)
#pragma unroll
            for (int r = 0; r < 8; ++r)
                qS[(lane + 32 * r) * LDS_P + w] = expf(av[r] - m);
        }
        // prefetch next step's feat slab (16 cols x 256 tags) while WMMAs run
        if (s + 1 < S_LEN)
            __builtin_prefetch(feats + ((size_t)(s + 1) * BATCH + c0) * TAGS + tid * 8, 0, 1);
        __syncthreads();

        // (c) C(16x16) = E_rows(16x256) @ Q(256x16): 64 f32 WMMAs, two
        //     accumulators, and a 4-deep software pipeline on B fragments so
        //     each WMMA waits with dscnt slack ~3 instead of 0.
        v8f acc0 = {}; v8f acc1 = {};
        {
            // B fragment (4x16): lane = N (=mrow); VGPR0: K = 2*half, VGPR1: K = 2*half+1
            const float* qb = qS + (2 * half) * LDS_P + mrow;
            v2f bf[4];
#pragma unroll
            for (int p = 0; p < 4; ++p) {
                v2f b;
                b.x = qb[(4 * p + 0) * LDS_P];
                b.y = qb[(4 * p + 1) * LDS_P];
                bf[p] = b;
            }
#pragma unroll
            for (int k = 0; k < 64; ++k) {
                if (k & 1)
                    acc1 = __builtin_amdgcn_wmma_f32_16x16x4_f32(
                               false, a[k], false, bf[k & 3], (short)0, acc1, false, false);
                else
                    acc0 = __builtin_amdgcn_wmma_f32_16x16x4_f32(
                               false, a[k], false, bf[k & 3], (short)0, acc0, false, false);
                if (k + 4 < 64) {
                    v2f b;
                    b.x = qb[(4 * (k + 4) + 0) * LDS_P];
                    b.y = qb[(4 * (k + 4) + 1) * LDS_P];
                    bf[k & 3] = b;
                }
            }
        }
        v8f acc = acc0 + acc1;

        // (d) new_alpha = log(C) + colmax + feat; masked blend; write back.
        //     C/D layout: element v at lane l -> M = v + 8*half, N = l&15.
        {
            const int   col = mrow;
            const float cm  = colMax[col];
            const float mk  = mask[(size_t)s * BATCH + c0 + col];
            const float omk = 1.0f - mk;
            const float* fb = feats + ((size_t)s * BATCH + c0 + col) * TAGS
                              + 16 * w + 8 * half;
            v4f f0 = *(const v4f*)(fb);
            v4f f1 = *(const v4f*)(fb + 4);
            float fv[8] = { f0.x, f0.y, f0.z, f0.w, f1.x, f1.y, f1.z, f1.w };
#pragma unroll
            for (int v = 0; v < 8; ++v) {
                int   tag = 16 * w + v + 8 * half;
                float old = alphaS[tag * LDS_P + col];
                float nv  = logf(acc[v]) + cm + fv[v];
                alphaS[tag * LDS_P + col] = mk * nv + omk * old;
            }
        }
        __syncthreads();
    }

    // ---- final: out[b] = LSE_j(alpha[j][b] + transition[END_TAG=1][j])
    {
        const int col = w;
        float m = -3.0e38f, sum = 0.0f;
        for (int j = lane; j < TAGS; j += 32) {
            float x  = alphaS[j * LDS_P + col] + trans[TAGS + j]; // row END_TAG=1
            float mn = fmaxf(m, x);
            sum = sum * expf(m - mn) + expf(x - mn);
            m = mn;
        }
#pragma unroll
        for (int off = 16; off > 0; off >>= 1) {
            float mo = __shfl_xor(m, off, 32);
            float so = __shfl_xor(sum, off, 32);
            float mn = fmaxf(m, mo);
            sum = sum * expf(m - mn) + so * expf(mo - mn);
            m = mn;
        }
        if (lane == 0) out[c0 + col] = logf(sum) + m;
    }
}

extern "C" void kernel_launch(void* const* d_in, const int* in_sizes, int n_in,
                              void* d_out, int out_size, void* d_ws, size_t ws_size,
                              hipStream_t stream)
{
    (void)in_sizes; (void)n_in; (void)d_ws; (void)ws_size; (void)out_size;
    const float* feats = (const float*)d_in[0];   // [128,64,256] f32
    const float* mask  = (const float*)d_in[1];   // [128,64]     f32
    const float* trans = (const float*)d_in[2];   // [256,256]    f32
    float* out = (float*)d_out;                   // [64]         f32

    crf_fwd_wmma_kernel<<<dim3(BATCH / COLS), dim3(NTHREAD), 0, stream>>>(
        feats, mask, trans, out);
}